// ResGCN_42314017800849
// MI455X (gfx1250) — compile-verified
//
#include <hip/hip_runtime.h>
#include <hip/hip_bf16.h>

// ResGCN layer: support = x @ W; agg = scatter_add(support[src]*a -> dst);
// out = relu(agg + bias + y)
//
// Phase 1: fp32 WMMA GEMM (V_WMMA_F32_16X16X4_F32). W staged in LDS in a
//          K-quad interleaved layout so each B fragment is a single aligned
//          ds_load_b64 (conflict-free: lanes 0-15 hit banks {0,1} mod 4,
//          lanes 16-31 hit banks {2,3} mod 4).
// Phase 2: one wave32 per edge; lane l handles float4 at column 4l
//          (coalesced 512B row read from L2-resident support),
//          4x global_atomic_add_f32 into d_out (also L2-resident).
// Phase 3: in-place relu(out + bias + y), float4 vectorized.

typedef __attribute__((ext_vector_type(2))) float v2f;
typedef __attribute__((ext_vector_type(8))) float v8f;

#define D 128  // D_IN == D_OUT == 128

// ---------------------------------------------------------------------------
// Phase 1: support[n,128] = x[n,128] @ W[128,128]   (fp32 WMMA)
// Block = 256 threads = 8 waves; block M-tile = 128 rows, wave M-tile = 16.
// LDS layout: word[(k>>2)*512 + 4*n + (k&3)] = W[k][n]
//   -> B frag for K-step s, lane l: one ds_load_b64 at byte offset
//      2048*s + 256*t + 16*l15 + 8*half
// ---------------------------------------------------------------------------
__global__ __launch_bounds__(256) void resgcn_gemm(
    const float* __restrict__ x, const float* __restrict__ w,
    float* __restrict__ support, int n_nodes) {
  __shared__ float wlds[D * D];  // 64 KB

  // Cooperative staging of W into the K-quad interleaved layout.
  #pragma unroll
  for (int i = threadIdx.x; i < (D * D) / 4; i += 256) {
    const float4 v = ((const float4*)w)[i];
    const int k = i >> 5;           // row index (i*4 / 128)
    const int n0 = (i & 31) << 2;   // column base
    float* p = wlds + (k >> 2) * 512 + (k & 3) + 4 * n0;
    p[0] = v.x;
    p[4] = v.y;
    p[8] = v.z;
    p[12] = v.w;
  }
  __syncthreads();

  const int wave = threadIdx.x >> 5;
  const int lane = threadIdx.x & 31;
  const int half = lane >> 4;   // 0: K = 4s+{0,1}; 1: K = 4s+{2,3}
  const int l15  = lane & 15;
  const int m0   = blockIdx.x * 128 + wave * 16;

  v8f acc[8] = {};  // 8 N-tiles of 16x16 f32 accumulators

  // A-matrix row for this lane (clamped for the ragged tail; stores guarded).
  int arow = m0 + l15;
  if (arow >= n_nodes) arow = n_nodes - 1;
  const float* xrow = x + (long)arow * D + 2 * half;

  // Per-lane loop-invariant part of the B address (words).
  const float* bbase = wlds + 4 * l15 + 2 * half;

  #pragma unroll 4
  for (int s = 0; s < D / 4; ++s) {
    // A frag: 16x4 f32 -> 2 VGPRs: v0 = A[M=l15][kb], v1 = A[M=l15][kb+1]
    const v2f a = *(const v2f*)(xrow + 4 * s);
    const float* brow = bbase + s * 512;
    #pragma unroll
    for (int t = 0; t < 8; ++t) {
      // B frag: {W[kb][16t+l15], W[kb+1][16t+l15]} as one aligned b64 load.
      const v2f b = *(const v2f*)(brow + t * 64);
#if __has_builtin(__builtin_amdgcn_wmma_f32_16x16x4_f32)
      acc[t] = __builtin_amdgcn_wmma_f32_16x16x4_f32(
          false, a, false, b, (short)0, acc[t], false, false);
#else
      acc[t][0] += a.x * b.x + a.y * b.y;  // compile-safety fallback only
#endif
    }
  }

  // Store D: VGPR r, lanes 0-15 -> row m0+r, lanes 16-31 -> row m0+r+8.
  #pragma unroll
  for (int t = 0; t < 8; ++t) {
    #pragma unroll
    for (int r = 0; r < 8; ++r) {
      const int m = m0 + r + 8 * half;
      if (m < n_nodes) support[(long)m * D + t * 16 + l15] = acc[t][r];
    }
  }
}

// ---------------------------------------------------------------------------
// Phase 2: scatter-add.  One wave per edge; lane l owns columns [4l, 4l+3].
// support (51.2 MB) and the accumulator (51.2 MB) both fit in the 192 MB L2,
// so the gather + atomic phase is L2-resident.
// ---------------------------------------------------------------------------
__global__ __launch_bounds__(256) void resgcn_scatter(
    const float* __restrict__ support, const int* __restrict__ esrc,
    const int* __restrict__ edst, const float* __restrict__ avals,
    float* __restrict__ agg, int n_edges) {
  const int lane = threadIdx.x & 31;
  const int wave_id = (blockIdx.x * blockDim.x + threadIdx.x) >> 5;
  const int n_waves = (gridDim.x * blockDim.x) >> 5;

  for (int e = wave_id; e < n_edges; e += n_waves) {
    const int s = esrc[e];      // uniform within wave -> single broadcast load
    const int d = edst[e];
    const float a = avals[e];
    const float4 v = ((const float4*)(support + (long)s * D))[lane];
    float* o = agg + (long)d * D + lane * 4;
    unsafeAtomicAdd(o + 0, v.x * a);   // global_atomic_add_f32 (no return)
    unsafeAtomicAdd(o + 1, v.y * a);
    unsafeAtomicAdd(o + 2, v.z * a);
    unsafeAtomicAdd(o + 3, v.w * a);
  }
}

// ---------------------------------------------------------------------------
// Phase 3: out = relu(out + bias + y), float4 vectorized, in place.
// ---------------------------------------------------------------------------
__global__ __launch_bounds__(256) void resgcn_epilogue(
    float* __restrict__ out, const float* __restrict__ y,
    const float* __restrict__ bias, int total4) {
  const int i = blockIdx.x * blockDim.x + threadIdx.x;
  if (i >= total4) return;
  const float4 o = ((const float4*)out)[i];
  const float4 yy = ((const float4*)y)[i];
  const float4 b = *(const float4*)(bias + ((i & 31) << 2));
  float4 r;
  r.x = fmaxf(o.x + b.x + yy.x, 0.0f);
  r.y = fmaxf(o.y + b.y + yy.y, 0.0f);
  r.z = fmaxf(o.z + b.z + yy.z, 0.0f);
  r.w = fmaxf(o.w + b.w + yy.w, 0.0f);
  ((float4*)out)[i] = r;
}

// ---------------------------------------------------------------------------
extern "C" void kernel_launch(void* const* d_in, const int* in_sizes, int n_in,
                              void* d_out, int out_size, void* d_ws,
                              size_t ws_size, hipStream_t stream) {
  const float* x     = (const float*)d_in[0];
  const float* y     = (const float*)d_in[1];
  const int*   esrc  = (const int*)d_in[2];
  const int*   edst  = (const int*)d_in[3];
  const float* avals = (const float*)d_in[4];
  const float* w     = (const float*)d_in[5];
  const float* bias  = (const float*)d_in[6];

  const int n_nodes = in_sizes[0] / D;   // 100000
  const int n_edges = in_sizes[2];       // 1600000

  float* out = (float*)d_out;
  float* support = (float*)d_ws;         // 100000*128*4 = 51.2 MB scratch

  // Zero the accumulator (d_out) every call -> deterministic, capture-safe.
  hipMemsetAsync(d_out, 0, (size_t)n_nodes * D * sizeof(float), stream);

  const int gemm_blocks = (n_nodes + 127) / 128;
  resgcn_gemm<<<gemm_blocks, 256, 0, stream>>>(x, w, support, n_nodes);

  resgcn_scatter<<<4096, 256, 0, stream>>>(support, esrc, edst, avals, out,
                                           n_edges);

  const int total4 = n_nodes * (D / 4);
  resgcn_epilogue<<<(total4 + 255) / 256, 256, 0, stream>>>(out, y, bias,
                                                            total4);
}